// MinimalLSTM_309237645636
// MI455X (gfx1250) — compile-verified
//
#include <hip/hip_runtime.h>

typedef __attribute__((ext_vector_type(2))) float v2f;
typedef __attribute__((ext_vector_type(4))) float v4f;
typedef __attribute__((ext_vector_type(8))) float v8f;

#define BQ 512
#define SQ 2048
#define EQ 20

__device__ __forceinline__ float sigm(float x) {
  return 1.0f / (1.0f + __expf(-x));
}
__device__ __forceinline__ float tanh_f(float x) {
  return 2.0f / (1.0f + __expf(-2.0f * x)) - 1.0f;
}

// D = A(16x4 f32) * B(4x16 f32) + C(16x16 f32), wave32.
// A: lane m(0-15): v0=A[m][0], v1=A[m][1]; lanes 16-31: K=2,3.
// B: v0: K=(lane<16?0:2) for N=lane%16; v1: K=(lane<16?1:3).
// C/D: lane holds column N=lane%16, rows (lane<16?0-7:8-15) in v0..v7.
__device__ __forceinline__ v8f wmma_16x16x4(v2f a, v2f b, v8f c) {
  return __builtin_amdgcn_wmma_f32_16x16x4_f32(false, a, false, b, (short)0, c,
                                               false, false);
}

// ---------------------------------------------------------------------------
// Phase 1: xin[sc][bt][lane][0..7] = (W_ih @ emb[x[b,s]]) + b_ih + b_hh
// One wave per (step-in-chunk, batch-tile). Gates on M axis, batch on N axis.
// ---------------------------------------------------------------------------
__global__ __launch_bounds__(256) void lstm_xin_kernel(
    const int* __restrict__ x, const float* __restrict__ emb,
    const float* __restrict__ W_ih, const float* __restrict__ b_ih,
    const float* __restrict__ b_hh, float* __restrict__ xin,
    int s0, int steps)
{
  const int lane = threadIdx.x & 31;
  const int wave = blockIdx.x * (blockDim.x >> 5) + (threadIdx.x >> 5);
  if (wave >= steps * 32) return;
  const int sc = wave >> 5;     // step within chunk
  const int bt = wave & 31;     // batch tile (16 rows each)
  const int s  = s0 + sc;
  const int n  = lane & 15;     // M-row (gate) for A, N-col (batch) for B/C/D
  const bool lo = lane < 16;
  const int koff = lo ? 0 : 2;

  // A chunks: W_ih[:, 4k:4k+4]  (gate row = n)
  v2f a[5];
#pragma unroll
  for (int k = 0; k < 5; ++k)
    a[k] = *(const v2f*)(W_ih + n * EQ + 4 * k + koff);

  // C initialized with combined bias (gate = vgpr + (lo?0:8))
  const int gbase = lo ? 0 : 8;
  v4f bi0 = *(const v4f*)(b_ih + gbase);
  v4f bi1 = *(const v4f*)(b_ih + gbase + 4);
  v4f bh0 = *(const v4f*)(b_hh + gbase);
  v4f bh1 = *(const v4f*)(b_hh + gbase + 4);
  v8f acc;
#pragma unroll
  for (int r = 0; r < 4; ++r) {
    acc[r]     = bi0[r] + bh0[r];
    acc[r + 4] = bi1[r] + bh1[r];
  }

  const int b = bt * 16 + n;
  const int tok = x[b * SQ + s];
  const float* erow = emb + (size_t)tok * EQ + koff;
#pragma unroll
  for (int k = 0; k < 5; ++k) {
    v2f bb = *(const v2f*)(erow + 4 * k);   // e[tok][4k+koff], e[tok][4k+koff+1]
    acc = wmma_16x16x4(a[k], bb, acc);
  }

  float* dst = xin + (((size_t)sc * 32 + bt) * 32 + lane) * 8;
  *(v8f*)dst = acc;
}

// ---------------------------------------------------------------------------
// Phase 2: sequential recurrence. One wave per 16-row batch tile.
// gates^T[m][n] = W_hh(16x4) x h^T(4x16) + xin-tile  via one WMMA per step.
// ---------------------------------------------------------------------------
__global__ __launch_bounds__(32) void lstm_rec_kernel(
    const float* __restrict__ xin, const float* __restrict__ W_hh,
    float* __restrict__ state, const float* __restrict__ W_fc,
    const float* __restrict__ b_fc, float* __restrict__ out,
    int steps, int first, int last)
{
  const int lane = threadIdx.x & 31;
  const int bt = blockIdx.x;          // 32 tiles
  const int n = lane & 15;            // batch column
  const bool lo = lane < 16;
  const int koff = lo ? 0 : 2;
  const int b = bt * 16 + n;

  const v2f a = *(const v2f*)(W_hh + n * 4 + koff);  // A = W_hh, gate row = n

  // Each lane keeps all 4 (h,c) for its batch column (redundant across halves
  // so the next step's B-matrix needs no shuffle).
  float h[4], c[4];
  if (first) {
#pragma unroll
    for (int j = 0; j < 4; ++j) { h[j] = 0.0f; c[j] = 0.0f; }
  } else {
#pragma unroll
    for (int j = 0; j < 4; ++j) {
      h[j] = state[b * 8 + j];
      c[j] = state[b * 8 + 4 + j];
    }
  }

  for (int sc = 0; sc < steps; ++sc) {
    const float* cp = xin + (((size_t)sc * 32 + bt) * 32 + lane) * 8;
    v8f cacc = *(const v8f*)cp;

    v2f bb;                       // B = h^T: v0 -> h[n][koff], v1 -> h[n][koff+1]
    bb.x = lo ? h[0] : h[2];
    bb.y = lo ? h[1] : h[3];

    v8f d = wmma_16x16x4(a, bb, cacc);

    // Swap lane halves: lo lanes hold gate rows 0-7 (i,f), hi hold 8-15 (g,o).
    float sw[8];
#pragma unroll
    for (int r = 0; r < 8; ++r) sw[r] = __shfl_xor(d[r], 16, 32);

#pragma unroll
    for (int j = 0; j < 4; ++j) {
      float ig = lo ? d[j]      : sw[j];       // gate row j
      float fg = lo ? d[j + 4]  : sw[j + 4];   // gate row j+4
      float gg = lo ? sw[j]     : d[j];        // gate row j+8
      float og = lo ? sw[j + 4] : d[j + 4];    // gate row j+12
      c[j] = sigm(fg) * c[j] + sigm(ig) * tanh_f(gg);
      h[j] = sigm(og) * tanh_f(c[j]);
    }
  }

  if (lo) {
#pragma unroll
    for (int j = 0; j < 4; ++j) {
      state[b * 8 + j]     = h[j];
      state[b * 8 + 4 + j] = c[j];
    }
    if (last) {
#pragma unroll
      for (int cc = 0; cc < 5; ++cc) {
        float acc = b_fc[cc];
#pragma unroll
        for (int j = 0; j < 4; ++j) acc += h[j] * W_fc[cc * 4 + j];
        out[b * 5 + cc] = acc;
      }
    }
  }
}

// ---------------------------------------------------------------------------
extern "C" void kernel_launch(void* const* d_in, const int* in_sizes, int n_in,
                              void* d_out, int out_size, void* d_ws, size_t ws_size,
                              hipStream_t stream)
{
  (void)in_sizes; (void)n_in; (void)out_size;
  const int*   x    = (const int*)d_in[0];
  const float* emb  = (const float*)d_in[1];
  const float* W_ih = (const float*)d_in[2];
  const float* W_hh = (const float*)d_in[3];
  const float* b_ih = (const float*)d_in[4];
  const float* b_hh = (const float*)d_in[5];
  const float* W_fc = (const float*)d_in[6];
  const float* b_fc = (const float*)d_in[7];
  float* out = (float*)d_out;

  // ws layout: [ h/c state : 16KB ][ xin chunk : SC * 32KB ]
  const size_t stateBytes   = (size_t)BQ * 8 * sizeof(float);
  float* state = (float*)d_ws;
  float* xin   = (float*)((char*)d_ws + stateBytes);
  size_t avail = (ws_size > stateBytes) ? ws_size - stateBytes : 0;
  const size_t bytesPerStep = (size_t)BQ * 16 * sizeof(float);   // 32 KB
  int SC = (int)(avail / bytesPerStep);
  if (SC > SQ) SC = SQ;
  if (SC < 1)  SC = 1;

  for (int s0 = 0; s0 < SQ; s0 += SC) {
    const int steps  = (SQ - s0 < SC) ? (SQ - s0) : SC;
    const int tiles  = steps * 32;              // waves needed
    const int blocks = (tiles + 7) / 8;         // 8 waves per 256-thread block
    lstm_xin_kernel<<<blocks, 256, 0, stream>>>(x, emb, W_ih, b_ih, b_hh,
                                                xin, s0, steps);
    lstm_rec_kernel<<<32, 32, 0, stream>>>(xin, W_hh, state, W_fc, b_fc, out,
                                           steps, s0 == 0 ? 1 : 0,
                                           (s0 + steps >= SQ) ? 1 : 0);
  }
}